// LMCL_90555090468955
// MI455X (gfx1250) — compile-verified
//
#include <hip/hip_runtime.h>
#include <math.h>

// ---- problem constants (match reference) ----
#define SSN 40000
#define HHN 60000
#define NTOT (SSN + HHN)           // 100000
#define DD 128
#define LL 3
#define BB 1024
#define SSL_TEMP 0.2f

typedef __attribute__((ext_vector_type(2))) float v2f;
typedef __attribute__((ext_vector_type(8))) float v8f;

// ---------------- wave32 reductions ----------------
__device__ __forceinline__ float wave_sum(float v) {
    for (int off = 16; off > 0; off >>= 1) v += __shfl_xor(v, off, 32);
    return v;
}
__device__ __forceinline__ float wave_max(float v) {
    for (int off = 16; off > 0; off >>= 1) v = fmaxf(v, __shfl_xor(v, off, 32));
    return v;
}

// ---------------- GCN kernels ----------------
// ego = concat(syn, herb); acc = ego; neigh = 0
__global__ void k_init(const float* __restrict__ syn, const float* __restrict__ herb,
                       float* __restrict__ ego, float* __restrict__ acc,
                       float* __restrict__ neigh) {
    size_t i = (size_t)blockIdx.x * blockDim.x + threadIdx.x;
    if (i >= (size_t)NTOT * DD) return;
    float v = (i < (size_t)SSN * DD) ? syn[i] : herb[i - (size_t)SSN * DD];
    ego[i] = v;
    acc[i] = v;
    neigh[i] = 0.0f;
}

// COO spmm: one wave per edge; lane moves float4 (32*16B = 512B = one D=128 row).
// Node matrix (51.2MB) is L2-resident on MI455X (192MB L2) -> on-chip gather/scatter.
__global__ void k_spmm(const int* __restrict__ rows, const int* __restrict__ cols,
                       const float* __restrict__ vals, const float* __restrict__ x,
                       float* __restrict__ y, int nnz) {
    size_t gtid = (size_t)blockIdx.x * blockDim.x + threadIdx.x;
    int e = (int)(gtid >> 5);
    int lane = threadIdx.x & 31;
    if (e >= nnz) return;
    int r = rows[e];
    int c = cols[e];
    float v = vals[e];
    float4 t = ((const float4*)(x + (size_t)c * DD))[lane];
    float* o = y + (size_t)r * DD + lane * 4;
    atomicAdd(o + 0, v * t.x);
    atomicAdd(o + 1, v * t.y);
    atomicAdd(o + 2, v * t.z);
    atomicAdd(o + 3, v * t.w);
}

// ego = relu(a*neigh + (1-a)*ego); acc += ego; neigh = 0 (ready for next layer)
__global__ void k_combine(float* __restrict__ ego, float* __restrict__ neigh,
                          float* __restrict__ acc, const float* __restrict__ alpha) {
    size_t i = (size_t)blockIdx.x * blockDim.x + threadIdx.x;
    if (i >= (size_t)NTOT * DD) return;
    float a = 1.0f / (1.0f + __expf(-alpha[0]));
    float v = fmaxf(a * neigh[i] + (1.0f - a) * ego[i], 0.0f);
    ego[i] = v;
    acc[i] += v;
    neigh[i] = 0.0f;
}

// mean = acc/4 -> out (adj pass writes straight into u_emb|i_emb output region)
__global__ void k_scale025(const float* __restrict__ acc, float* __restrict__ out) {
    size_t i = (size_t)blockIdx.x * blockDim.x + threadIdx.x;
    if (i >= (size_t)NTOT * DD) return;
    out[i] = acc[i] * 0.25f;
}

// z = mean / ||mean|| = acc / ||acc|| (0.25 scale cancels). One wave per row.
__global__ void k_norm(const float* __restrict__ acc, float* __restrict__ z) {
    int r = blockIdx.x;
    int lane = threadIdx.x;
    float4 t = ((const float4*)(acc + (size_t)r * DD))[lane];
    float ss = t.x * t.x + t.y * t.y + t.z * t.z + t.w * t.w;
    ss = wave_sum(ss);
    float inv = rsqrtf(ss);
    float4 o;
    o.x = t.x * inv; o.y = t.y * inv; o.z = t.z * inv; o.w = t.w * inv;
    ((float4*)(z + (size_t)r * DD))[lane] = o;
}

// dst[b,:] = src[idx[b]+rowBase, :]   (block of 128 threads, one row per block)
__global__ void k_gather(float* __restrict__ dst, const float* __restrict__ src,
                         const int* __restrict__ idx, int rowBase) {
    int b = blockIdx.x;
    int t = threadIdx.x;
    dst[(size_t)b * DD + t] = src[(size_t)(idx[b] + rowBase) * DD + t];
}

// pos[b] = dot(A1[b], A2[b]) — one wave per b
__global__ void k_pos(const float* __restrict__ A1, const float* __restrict__ A2,
                      float* __restrict__ pos) {
    int b = blockIdx.x;
    int lane = threadIdx.x;
    float4 x = ((const float4*)(A1 + (size_t)b * DD))[lane];
    float4 y = ((const float4*)(A2 + (size_t)b * DD))[lane];
    float s = wave_sum(x.x * y.x + x.y * y.y + x.z * y.z + x.w * y.w);
    if (lane == 0) pos[b] = s;
}

// sup_logits[b] = <u[users[b]], i[items[b]] - i[neg[b]]> — one wave per b
__global__ void k_sup(const float* __restrict__ uemb, const float* __restrict__ iemb,
                      const int* __restrict__ users, const int* __restrict__ items,
                      const int* __restrict__ negs, float* __restrict__ out) {
    int b = blockIdx.x;
    int lane = threadIdx.x;
    float4 u = ((const float4*)(uemb + (size_t)users[b] * DD))[lane];
    float4 p = ((const float4*)(iemb + (size_t)items[b] * DD))[lane];
    float4 n = ((const float4*)(iemb + (size_t)negs[b] * DD))[lane];
    float s = wave_sum(u.x * (p.x - n.x) + u.y * (p.y - n.y) +
                       u.z * (p.z - n.z) + u.w * (p.w - n.w));
    if (lane == 0) out[b] = s;
}

// ---------------- WMMA fp32 GEMM, M-register-blocked ----------------
// out[m, n] = scale * dot(A[m,:], Bm[n,:]) - (bias ? bias[m] : 0)
// A: [M, 128] row-major (M = 64*gridDim.y); Bm: [16*ntiles, 128] row-major.
// Each wave computes a 64x16 output strip: 4 accumulators along M share every
// B fragment (4x less B traffic; loop body = 5 b64 loads : 4 WMMAs).
// fp32 A 16x4 fragment: lanes0-15 M=lane K={0,1}; lanes16-31 M=lane-16 K={2,3}.
// fp32 B 4x16 fragment mirrored; C/D: vgpr i -> M=i (+8 for lanes16-31), N=lane&15.
__global__ void k_gemm_wmma(const float* __restrict__ A, const float* __restrict__ Bm,
                            const float* __restrict__ bias, float* __restrict__ out,
                            int ntiles, int ldo, float scale) {
    int wave = threadIdx.x >> 5;
    int ntile = blockIdx.x * 4 + wave;
    int mbase = blockIdx.y * 64;
    if (ntile >= ntiles) return;           // wave-uniform: EXEC stays all-ones for WMMA
    int lane = threadIdx.x & 31;
    int mr = lane & 15;
    int kp = (lane >> 4) << 1;             // 0 or 2: this lane's K-pair within the 4-step
    const float* A0 = A + (size_t)(mbase + mr) * DD;
    const float* Brow = Bm + (size_t)(ntile * 16 + mr) * DD;
    v8f c0 = {0.f, 0.f, 0.f, 0.f, 0.f, 0.f, 0.f, 0.f};
    v8f c1 = c0, c2 = c0, c3 = c0;
#pragma unroll
    for (int k0 = 0; k0 < DD; k0 += 4) {
        v2f b  = *(const v2f*)(Brow + k0 + kp);            // 8B-aligned b64 loads
        v2f a0 = *(const v2f*)(A0 + 0 * 16 * DD + k0 + kp);
        v2f a1 = *(const v2f*)(A0 + 1 * 16 * DD + k0 + kp);
        v2f a2 = *(const v2f*)(A0 + 2 * 16 * DD + k0 + kp);
        v2f a3 = *(const v2f*)(A0 + 3 * 16 * DD + k0 + kp);
        c0 = __builtin_amdgcn_wmma_f32_16x16x4_f32(false, a0, false, b, (short)0, c0,
                                                   false, false);
        c1 = __builtin_amdgcn_wmma_f32_16x16x4_f32(false, a1, false, b, (short)0, c1,
                                                   false, false);
        c2 = __builtin_amdgcn_wmma_f32_16x16x4_f32(false, a2, false, b, (short)0, c2,
                                                   false, false);
        c3 = __builtin_amdgcn_wmma_f32_16x16x4_f32(false, a3, false, b, (short)0, c3,
                                                   false, false);
    }
    int cm = (lane >> 4) << 3;             // lanes 16-31 hold rows M+8
    int cn = lane & 15;
    float* ob = out + (size_t)ntile * 16 + cn;
#pragma unroll
    for (int j = 0; j < 4; ++j) {
        v8f c = (j == 0) ? c0 : (j == 1) ? c1 : (j == 2) ? c2 : c3;
#pragma unroll
        for (int i = 0; i < 8; ++i) {
            int m = mbase + j * 16 + i + cm;
            float r = c[i] * scale - (bias ? bias[m] : 0.0f);
            ob[(size_t)m * ldo] = r;
        }
    }
}

// ---------------- InfoNCE row reduction ----------------
// loss_r = logsumexp(logits[r,:]) - logits[r,r]; atomicAdd(mean contribution)
__global__ void k_nce(const float* __restrict__ logits, float* __restrict__ out_scalar) {
    int r = blockIdx.x;
    int lane = threadIdx.x;
    float mx = -3.4e38f;
    for (int c = lane; c < BB; c += 32) mx = fmaxf(mx, logits[(size_t)r * BB + c]);
    mx = wave_max(mx);
    float s = 0.0f;
    for (int c = lane; c < BB; c += 32) s += __expf(logits[(size_t)r * BB + c] - mx);
    s = wave_sum(s);
    if (lane == 0) {
        float lse = mx + __logf(s);
        float loss = lse - logits[(size_t)r * BB + r];
        atomicAdd(out_scalar, loss * (1.0f / (float)BB));
    }
}

__global__ void k_zero1(float* __restrict__ p) {
    if (threadIdx.x == 0 && blockIdx.x == 0) *p = 0.0f;
}

// ---------------- host orchestration ----------------
extern "C" void kernel_launch(void* const* d_in, const int* in_sizes, int n_in,
                              void* d_out, int out_size, void* d_ws, size_t ws_size,
                              hipStream_t stream) {
    (void)n_in; (void)out_size; (void)ws_size;
    const float* syn  = (const float*)d_in[0];
    const float* herb = (const float*)d_in[1];
    const int*   ar   = (const int*)d_in[2];
    const int*   ac   = (const int*)d_in[3];
    const float* av   = (const float*)d_in[4];
    const int*   r1   = (const int*)d_in[5];
    const int*   c1   = (const int*)d_in[6];
    const float* v1   = (const float*)d_in[7];
    const int*   r2   = (const int*)d_in[8];
    const int*   c2   = (const int*)d_in[9];
    const float* v2   = (const float*)d_in[10];
    const int*   users = (const int*)d_in[11];
    const int*   items = (const int*)d_in[12];
    const int*   negs  = (const int*)d_in[13];
    const float* alpha = (const float*)d_in[14];
    int nnzA = in_sizes[2], nnz1 = in_sizes[5], nnz2 = in_sizes[8];

    float* out = (float*)d_out;
    const size_t O_SUP  = 0;
    const size_t O_SSLU = O_SUP + BB;
    const size_t O_SSLI = O_SSLU + (size_t)BB * SSN;
    const size_t O_UEMB = O_SSLI + (size_t)BB * HHN;
    const size_t O_IEMB = O_UEMB + (size_t)SSN * DD;     // contiguous after u_emb
    const size_t O_NCE  = O_IEMB + (size_t)HHN * DD;

    const size_t ND = (size_t)NTOT * DD;
    const size_t BD = (size_t)BB * DD;
    float* ws    = (float*)d_ws;
    float* ego   = ws;
    float* neigh = ego + ND;
    float* acc   = neigh + ND;
    float* z1    = acc + ND;
    float* z2    = z1 + ND;
    float* A1u   = z2 + ND;
    float* A2u   = A1u + BD;
    float* A1i   = A2u + BD;
    float* A2i   = A1i + BD;
    float* posu  = A2i + BD;
    float* posi  = posu + BB;
    float* logits = posi + BB;

    auto cdiv = [](size_t a, size_t b) { return (unsigned)((a + b - 1) / b); };
    dim3 ewise(cdiv(ND, 256));

    auto gcn = [&](const int* rr, const int* cc, const float* vv, int nnz) {
        k_init<<<ewise, 256, 0, stream>>>(syn, herb, ego, acc, neigh);
        for (int l = 0; l < LL; ++l) {
            k_spmm<<<cdiv((size_t)nnz * 32, 256), 256, 0, stream>>>(rr, cc, vv, ego, neigh, nnz);
            k_combine<<<ewise, 256, 0, stream>>>(ego, neigh, acc, alpha);
        }
    };

    // main adjacency: mean embedding directly into output (u_emb|i_emb contiguous)
    gcn(ar, ac, av, nnzA);
    k_scale025<<<ewise, 256, 0, stream>>>(acc, out + O_UEMB);
    // subgraph views: l2-normalized means
    gcn(r1, c1, v1, nnz1);
    k_norm<<<NTOT, 32, 0, stream>>>(acc, z1);
    gcn(r2, c2, v2, nnz2);
    k_norm<<<NTOT, 32, 0, stream>>>(acc, z2);

    // batch gathers
    k_gather<<<BB, 128, 0, stream>>>(A1u, z1, users, 0);
    k_gather<<<BB, 128, 0, stream>>>(A2u, z2, users, 0);
    k_gather<<<BB, 128, 0, stream>>>(A1i, z1, items, SSN);
    k_gather<<<BB, 128, 0, stream>>>(A2i, z2, items, SSN);
    k_pos<<<BB, 32, 0, stream>>>(A1u, A2u, posu);
    k_pos<<<BB, 32, 0, stream>>>(A1i, A2i, posi);
    k_sup<<<BB, 32, 0, stream>>>(out + O_UEMB, out + O_IEMB, users, items, negs, out + O_SUP);

    // ssl_user = A1u @ u2^T - pos_u   (u2 = z2 user rows [0,SS))
    { int nt = SSN / 16; dim3 g(cdiv(nt, 4), BB / 64);
      k_gemm_wmma<<<g, 128, 0, stream>>>(A1u, z2, posu, out + O_SSLU, nt, SSN, 1.0f); }
    // ssl_item = A1i @ i2^T - pos_i   (i2 = z2 rows [SS,N))
    { int nt = HHN / 16; dim3 g(cdiv(nt, 4), BB / 64);
      k_gemm_wmma<<<g, 128, 0, stream>>>(A1i, z2 + (size_t)SSN * DD, posi,
                                         out + O_SSLI, nt, HHN, 1.0f); }

    // infonce = nce(user, temp=0.2) + nce(item, temp=1.0)
    k_zero1<<<1, 32, 0, stream>>>(out + O_NCE);
    { int nt = BB / 16; dim3 g(cdiv(nt, 4), BB / 64);
      k_gemm_wmma<<<g, 128, 0, stream>>>(A1u, A2u, nullptr, logits, nt, BB,
                                         1.0f / SSL_TEMP); }
    k_nce<<<BB, 32, 0, stream>>>(logits, out + O_NCE);
    { int nt = BB / 16; dim3 g(cdiv(nt, 4), BB / 64);
      k_gemm_wmma<<<g, 128, 0, stream>>>(A1i, A2i, nullptr, logits, nt, BB, 1.0f); }
    k_nce<<<BB, 32, 0, stream>>>(logits, out + O_NCE);
}